// GTOutputWithPooling_71330816852702
// MI455X (gfx1250) — compile-verified
//
#include <hip/hip_runtime.h>
#include <hip/hip_bf16.h>

typedef __attribute__((ext_vector_type(2))) float v2f;
typedef __attribute__((ext_vector_type(8))) float v8f;

namespace {
constexpr int kB   = 8;
constexpr int kS   = 512;
constexpr int kG   = 16;
constexpr int kWin = 12;
constexpr int kDa  = 1024;
constexpr int kDm  = 256;
}

// One streaming pass per (b, d) column over S:
//  - cols [0,D):    gathered gap rows x[b, gap[b,g], d]
//  - cols [D,2D):   g==0 -> global col max; g>=1 -> windowed max of x*wmf
//                   (reference maxes over ALL s, so it is floored at 0)
//  - cols [2D,3D):  g==0 -> col sum / sum(mask); g>=1 -> masked window sum / count
__global__ void build_cat_kernel(const float* __restrict__ x,
                                 const int*   __restrict__ gap,
                                 const int*   __restrict__ mask,
                                 float*       __restrict__ cat,
                                 int D)
{
    const int blocksPerB = D / 256;
    const int b = blockIdx.x / blocksPerB;
    const int d = (blockIdx.x % blocksPerB) * 256 + threadIdx.x;
    const float* __restrict__ xb = x + (size_t)b * kS * D;
    const int*   __restrict__ mb = mask + b * kS;
    const int D3 = 3 * D;

    float colmax = -__builtin_inff();
    float colsum = 0.f;
    int   msum   = 0;
    for (int s = 0; s < kS; ++s) {
        float v = xb[(size_t)s * D + d];     // coalesced; mask load is scalar-uniform
        colmax = fmaxf(colmax, v);
        colsum += v;
        msum += (mb[s] > 0) ? 1 : 0;
    }

    // gap-row gathers -> cols [0, D)
    for (int g = 0; g < kG; ++g) {
        int gi = gap[b * kG + g];
        cat[(size_t)(b * kG + g) * D3 + d] = xb[(size_t)gi * D + d];
    }

    // row g = 0: global pools
    {
        size_t r = (size_t)(b * kG) * D3;
        cat[r + D + d]     = colmax;
        cat[r + 2 * D + d] = colsum / (float)msum;
    }

    // rows g = 1..15: windowed pools around gap_indices[b, g] (gaps_rest)
    for (int g = 1; g < kG; ++g) {
        int gp = gap[b * kG + g];
        int lo = gp - kWin; if (lo < 0)      lo = 0;
        int hi = gp + kWin; if (hi > kS - 1) hi = kS - 1;
        float wmax = 0.f;   // zeros outside the window participate in the ref max
        float wsum = 0.f;
        int   wcnt = 0;
        for (int s = lo; s <= hi; ++s) {     // re-reads hit L2 (whole tensor resident)
            if (mb[s] > 0) {
                float v = xb[(size_t)s * D + d];
                wmax = fmaxf(wmax, v);
                wsum += v;
                ++wcnt;
            }
        }
        size_t r = (size_t)(b * kG + g) * D3;
        cat[r + D + d]     = wmax;
        cat[r + 2 * D + d] = wsum / (float)wcnt;
    }
}

// Fused matvec: logits[b,g] = cat_att[b,g,:]·W_att + cat_mod[b,g,:]·W_mod + biases
// One wave per batch tile (rows 16t..16t+15), V_WMMA_F32_16X16X4_F32.
// A layout (16x4 f32): lanes 0-15 hold M=lane, K={0,1}; lanes 16-31 hold M=lane-16, K={2,3}.
// B (4x16): weights live in column 0 only -> lane 0 carries K={0,1}, lane 16 carries K={2,3}.
// Result column N=0: lane 0 has M=0..7 in c[0..7], lane 16 has M=8..15.
__global__ void __launch_bounds__(32)
logits_wmma_kernel(const float* __restrict__ att_cat,
                   const float* __restrict__ mod_cat,
                   const float* __restrict__ W_att,
                   const float* __restrict__ b_att,
                   const float* __restrict__ W_mod,
                   const float* __restrict__ b_mod,
                   float*       __restrict__ out)
{
    const int t    = blockIdx.x;      // tile == batch index
    const int lane = threadIdx.x;     // 0..31
    const int half = lane >> 4;       // K half select
    const int col  = lane & 15;       // A row within tile / B column

    v8f c0 = {};
    v8f c1 = {};   // second accumulator breaks the D->SRC2 dependency chain

    // ---- att_cat part: K = 3*1024 ----
    {
        const float* __restrict__ Arow = att_cat + (size_t)(t * kG + col) * (3 * kDa);
        for (int k = 0; k < 3 * kDa; k += 8) {
            const int k0 = k + half * 2;
            const int k1 = k + 4 + half * 2;
            v2f a0, a1, bv0, bv1;
            a0.x = Arow[k0];     a0.y = Arow[k0 + 1];
            a1.x = Arow[k1];     a1.y = Arow[k1 + 1];
            float w00 = W_att[k0], w01 = W_att[k0 + 1];
            float w10 = W_att[k1], w11 = W_att[k1 + 1];
            bv0.x = (col == 0) ? w00 : 0.f;
            bv0.y = (col == 0) ? w01 : 0.f;
            bv1.x = (col == 0) ? w10 : 0.f;
            bv1.y = (col == 0) ? w11 : 0.f;
            c0 = __builtin_amdgcn_wmma_f32_16x16x4_f32(false, a0, false, bv0,
                                                       (short)0, c0, false, false);
            c1 = __builtin_amdgcn_wmma_f32_16x16x4_f32(false, a1, false, bv1,
                                                       (short)0, c1, false, false);
        }
    }

    // ---- mod_cat part: K = 3*256 ----
    {
        const float* __restrict__ Arow = mod_cat + (size_t)(t * kG + col) * (3 * kDm);
        for (int k = 0; k < 3 * kDm; k += 8) {
            const int k0 = k + half * 2;
            const int k1 = k + 4 + half * 2;
            v2f a0, a1, bv0, bv1;
            a0.x = Arow[k0];     a0.y = Arow[k0 + 1];
            a1.x = Arow[k1];     a1.y = Arow[k1 + 1];
            float w00 = W_mod[k0], w01 = W_mod[k0 + 1];
            float w10 = W_mod[k1], w11 = W_mod[k1 + 1];
            bv0.x = (col == 0) ? w00 : 0.f;
            bv0.y = (col == 0) ? w01 : 0.f;
            bv1.x = (col == 0) ? w10 : 0.f;
            bv1.y = (col == 0) ? w11 : 0.f;
            c0 = __builtin_amdgcn_wmma_f32_16x16x4_f32(false, a0, false, bv0,
                                                       (short)0, c0, false, false);
            c1 = __builtin_amdgcn_wmma_f32_16x16x4_f32(false, a1, false, bv1,
                                                       (short)0, c1, false, false);
        }
    }

    const float bias = b_att[0] + b_mod[0];
    if (col == 0) {
        const int base = t * kG + half * 8;
        #pragma unroll
        for (int v = 0; v < 8; ++v)
            out[base + v] = c0[v] + c1[v] + bias;
    }
}

extern "C" void kernel_launch(void* const* d_in, const int* in_sizes, int n_in,
                              void* d_out, int out_size, void* d_ws, size_t ws_size,
                              hipStream_t stream) {
    const float* att   = (const float*)d_in[0];   // (8, 512, 1024)
    const float* mod   = (const float*)d_in[1];   // (8, 512, 256)
    const float* W_att = (const float*)d_in[2];   // (3072,)
    const float* b_att = (const float*)d_in[3];   // (1,)
    const float* W_mod = (const float*)d_in[4];   // (768,)
    const float* b_mod = (const float*)d_in[5];   // (1,)
    // d_in[6] = q_enc  (unused by the reference)
    const int*   gap   = (const int*)d_in[7];     // (8, 16)
    const int*   mask  = (const int*)d_in[8];     // (8, 512)
    // d_in[9] = q_mask (unused by the reference)
    float* out = (float*)d_out;                   // (8, 16)

    float* att_cat = (float*)d_ws;                              // 128 x 3072 f32
    float* mod_cat = att_cat + (size_t)kB * kG * 3 * kDa;       // 128 x 768  f32
    // total workspace: ~1.9 MB

    build_cat_kernel<<<kB * (kDa / 256), 256, 0, stream>>>(att, gap, mask, att_cat, kDa);
    build_cat_kernel<<<kB * (kDm / 256), 256, 0, stream>>>(mod, gap, mask, mod_cat, kDm);
    logits_wmma_kernel<<<kB, 32, 0, stream>>>(att_cat, mod_cat,
                                              W_att, b_att, W_mod, b_mod, out);
}